// Involution_19670950216065
// MI455X (gfx1250) — compile-verified
//
#include <hip/hip_runtime.h>
#include <cstdint>
#include <cstddef>

// Broadcast multiply: out[b][c][p] = x[b][p] * kernels[c][p], p = H*W pixels.
// Bandwidth-bound (~855 MB total traffic, ~37 us floor at 23.3 TB/s).
// CDNA5 paths used: global_load_async_to_lds_b128 (ASYNCcnt) to stage the
// kernel tile in LDS, ds_load_b128 reuse across 8 b-values, nontemporal
// b128 loads/stores for the streaming x / out traffic.

typedef float v4f __attribute__((ext_vector_type(4)));

#define TPB 256          // threads per block (8 waves of 32)
#define NC  32           // C dimension
#define BG  8            // b-values handled per block (kernel-tile reuse)

static constexpr int P  = 224 * 224;  // 50176 floats per plane
static constexpr int P4 = P / 4;      // 12544 float4 chunks per plane
static constexpr int PB = P * 4;      // plane stride in bytes (200704)

__global__ __launch_bounds__(TPB) void involution_bcast_mul(
    const float* __restrict__ x,
    const float* __restrict__ kern,
    float* __restrict__ out)
{
    __shared__ v4f tile[NC * TPB];                 // 128 KB LDS kernel tile

    const int tid  = threadIdx.x;
    const int pix4 = blockIdx.x * TPB + tid;       // float4 index within plane
    const uint32_t pbyte = (uint32_t)pix4 * 16u;   // byte offset within plane

    // ---- Stage 32 x TPB float4 kernel tile into LDS with async copies ----
    const char* kbase = (const char*)kern;
#pragma unroll
    for (int c = 0; c < NC; ++c) {
        // LDS byte address = low 32 bits of the flat shared address
        uint32_t lds  = (uint32_t)(uintptr_t)(void*)&tile[c * TPB + tid];
        uint32_t goff = (uint32_t)c * (uint32_t)PB + pbyte;
        asm volatile("global_load_async_to_lds_b128 %0, %1, %2"
                     :: "v"(lds), "v"(goff), "s"(kbase)
                     : "memory");
    }
    asm volatile("s_wait_asynccnt 0" ::: "memory");
    __syncthreads();

    const int b0 = blockIdx.y * BG;
    const char* xbase = (const char*)x;
    char*       obase = (char*)out;

#pragma unroll 1
    for (int bb = 0; bb < BG; ++bb) {
        const int b = b0 + bb;
        // x element used exactly once grid-wide at this (b, pixel): NT load
        v4f xv = __builtin_nontemporal_load(
            (const v4f*)(xbase + (size_t)b * PB + pbyte));
        char* ob = obase + ((size_t)b * NC) * (size_t)PB + pbyte;
#pragma unroll
        for (int c = 0; c < NC; ++c) {
            v4f kv = tile[c * TPB + tid];          // ds_load_b128
            v4f r  = xv * kv;
            // output written once, never re-read: NT store (keep L2 for kernels)
            __builtin_nontemporal_store(r, (v4f*)(ob + (size_t)c * PB));
        }
    }
}

extern "C" void kernel_launch(void* const* d_in, const int* in_sizes, int n_in,
                              void* d_out, int out_size, void* d_ws, size_t ws_size,
                              hipStream_t stream) {
    const float* x    = (const float*)d_in[0];   // [128, 224, 224] f32
    const float* kern = (const float*)d_in[1];   // [32, 224, 224] f32
    float*       out  = (float*)d_out;           // [128, 32, 224, 224] f32

    dim3 grid(P4 / TPB, 128 / BG);               // (49, 16)
    involution_bcast_mul<<<grid, TPB, 0, stream>>>(x, kern, out);
}